// Greed_38388417692531
// MI455X (gfx1250) — compile-verified
//
#include <hip/hip_runtime.h>

#define NNODES 100000
#define NEDGES 1600000
#define HID    64
#define INDIM  32
#define OUTDIM 32
#define NGRAPH 128
#define NLAYERS 4
#define POOLD  (HID * (NLAYERS + 1)) /* 320 */

typedef float v2f __attribute__((ext_vector_type(2)));
typedef float v8f __attribute__((ext_vector_type(8)));

__device__ __forceinline__ v8f wmma_f32(v2f a, v2f b, v8f c) {
  // D = A(16x4 f32) x B(4x16 f32) + C(16x16 f32)
  return __builtin_amdgcn_wmma_f32_16x16x4_f32(false, a, false, b, (short)0, c,
                                               false, false);
}

// ---------------------------------------------------------------------------
// Pre MLP: xcur = xres = x_in @ pre_w + pre_b ; pool emb[0] into gpool[:,0:64]
// One wave per 16-row tile, 8 waves per block.
// ---------------------------------------------------------------------------
__global__ __launch_bounds__(256) void pre_kernel(
    const float* __restrict__ xin, const float* __restrict__ w,
    const float* __restrict__ bias, const int* __restrict__ batch,
    float* __restrict__ xcur, float* __restrict__ xres, float* gpool) {
  __shared__ float s_w[INDIM * HID];     // 8 KB
  __shared__ float s_a[8][16 * INDIM];   // 16 KB
  const int tid = threadIdx.x, wave = tid >> 5, lane = tid & 31;
  const int mrow = lane & 15, kh = lane >> 4;

  for (int i = tid; i < INDIM * HID / 4; i += 256)
    ((float4*)s_w)[i] = ((const float4*)w)[i];

  const int row0 = (blockIdx.x * 8 + wave) * 16;
  for (int i = lane; i < 16 * INDIM / 4; i += 32) {
    int r = i >> 3, c4 = i & 7;
    int row = row0 + r; if (row > NNODES - 1) row = NNODES - 1;
    ((float4*)s_a[wave])[i] = ((const float4*)(xin + (size_t)row * INDIM))[c4];
  }
  __syncthreads();

  v8f acc[4];
#pragma unroll
  for (int j = 0; j < 4; ++j) {
    float b = bias[j * 16 + mrow];
#pragma unroll
    for (int v = 0; v < 8; ++v) acc[j][v] = b;
#pragma unroll
    for (int k = 0; k < INDIM / 4; ++k) {
      int kb = 4 * k + 2 * kh;
      v2f a = *(const v2f*)&s_a[wave][mrow * INDIM + kb];
      v2f bb;
      bb.x = s_w[kb * HID + j * 16 + mrow];
      bb.y = s_w[(kb + 1) * HID + j * 16 + mrow];
      acc[j] = wmma_f32(a, bb, acc[j]);
    }
  }

#pragma unroll
  for (int v = 0; v < 8; ++v) {
    int row = row0 + v + 8 * kh;
    if (row < NNODES) {
      int gofs = batch[row] * POOLD;
      size_t rb = (size_t)row * HID;
#pragma unroll
      for (int j = 0; j < 4; ++j) {
        int col = j * 16 + mrow;
        float val = acc[j][v];
        xcur[rb + col] = val;
        xres[rb + col] = val;
        atomicAdd(&gpool[gofs + col], val);
      }
    }
  }
}

// ---------------------------------------------------------------------------
// Edge aggregation: agg[dst] += x[src].  16 lanes per edge, float4 gather
// (one full 256B feature row per half-wave) + 4 f32 atomics per lane.
// Node features fit in L2 (192MB), so this runs at L2/atomic throughput.
// ---------------------------------------------------------------------------
__global__ __launch_bounds__(256) void scatter_kernel(
    const float* __restrict__ x, const int* __restrict__ ei,
    float* __restrict__ agg) {
  long long tid = (long long)blockIdx.x * 256 + threadIdx.x;
  int e = (int)(tid >> 4);
  if (e >= NEDGES) return;
  int part = (int)(tid & 15);
  int src = ei[e];
  int dst = ei[NEDGES + e];
  float4 v = ((const float4*)(x + (size_t)src * HID))[part];
  float* d = agg + (size_t)dst * HID + part * 4;
  atomicAdd(d + 0, v.x);
  atomicAdd(d + 1, v.y);
  atomicAdd(d + 2, v.z);
  atomicAdd(d + 3, v.w);
}

// ---------------------------------------------------------------------------
// Fused GIN layer: h = (x+agg); h1 = relu(h@w1+b1); h2 = h1@w2+b2;
// odd layer: h2 += xres, xres = h2;  x = relu(h2); pool into gpool.
// Two back-to-back WMMA GEMMs with LDS relayout between them.
// x is read AND written in place (tile-private rows, agg precomputed).
// ---------------------------------------------------------------------------
__global__ __launch_bounds__(256) void layer_kernel(
    float* x, const float* __restrict__ agg, const float* __restrict__ w1g,
    const float* __restrict__ b1g, const float* __restrict__ w2g,
    const float* __restrict__ b2g, const int* __restrict__ batch, float* xres,
    float* gpool, int layer) {
  __shared__ float s_w1[HID * HID];    // 16 KB
  __shared__ float s_w2[HID * HID];    // 16 KB
  __shared__ float s_a[8][16 * HID];   // 32 KB
  const int tid = threadIdx.x, wave = tid >> 5, lane = tid & 31;
  const int mrow = lane & 15, kh = lane >> 4;
  const float* w1 = w1g + (size_t)layer * HID * HID;
  const float* w2 = w2g + (size_t)layer * HID * HID;

  for (int i = tid; i < HID * HID / 4; i += 256) {
    ((float4*)s_w1)[i] = ((const float4*)w1)[i];
    ((float4*)s_w2)[i] = ((const float4*)w2)[i];
  }

  const int row0 = (blockIdx.x * 8 + wave) * 16;
  for (int i = lane; i < 16 * HID / 4; i += 32) {
    int r = i >> 4, c4 = i & 15;
    int row = row0 + r; if (row > NNODES - 1) row = NNODES - 1;
    float4 xv = ((const float4*)(x + (size_t)row * HID))[c4];
    float4 av = ((const float4*)(agg + (size_t)row * HID))[c4];
    float4 s;
    s.x = xv.x + av.x; s.y = xv.y + av.y; s.z = xv.z + av.z; s.w = xv.w + av.w;
    ((float4*)s_a[wave])[i] = s;
  }
  __syncthreads();

  v8f acc[4];
  // GEMM1: (x+agg) @ w1 + b1, relu
#pragma unroll
  for (int j = 0; j < 4; ++j) {
    float b = b1g[layer * HID + j * 16 + mrow];
#pragma unroll
    for (int v = 0; v < 8; ++v) acc[j][v] = b;
#pragma unroll
    for (int k = 0; k < HID / 4; ++k) {
      int kb = 4 * k + 2 * kh;
      v2f a = *(const v2f*)&s_a[wave][mrow * HID + kb];
      v2f bb;
      bb.x = s_w1[kb * HID + j * 16 + mrow];
      bb.y = s_w1[(kb + 1) * HID + j * 16 + mrow];
      acc[j] = wmma_f32(a, bb, acc[j]);
    }
  }
  __syncthreads();
  // D (row-striped-across-lanes) -> A layout (row-per-lane) via LDS, + relu
#pragma unroll
  for (int j = 0; j < 4; ++j)
#pragma unroll
    for (int v = 0; v < 8; ++v)
      s_a[wave][(v + 8 * kh) * HID + j * 16 + mrow] = fmaxf(acc[j][v], 0.f);
  __syncthreads();

  // GEMM2: h1 @ w2 + b2
#pragma unroll
  for (int j = 0; j < 4; ++j) {
    float b = b2g[layer * HID + j * 16 + mrow];
#pragma unroll
    for (int v = 0; v < 8; ++v) acc[j][v] = b;
#pragma unroll
    for (int k = 0; k < HID / 4; ++k) {
      int kb = 4 * k + 2 * kh;
      v2f a = *(const v2f*)&s_a[wave][mrow * HID + kb];
      v2f bb;
      bb.x = s_w2[kb * HID + j * 16 + mrow];
      bb.y = s_w2[(kb + 1) * HID + j * 16 + mrow];
      acc[j] = wmma_f32(a, bb, acc[j]);
    }
  }

  const bool res = (layer & 1) != 0;
#pragma unroll
  for (int v = 0; v < 8; ++v) {
    int row = row0 + v + 8 * kh;
    if (row < NNODES) {
      int gofs = batch[row] * POOLD + (layer + 1) * HID;
      size_t rb = (size_t)row * HID;
#pragma unroll
      for (int j = 0; j < 4; ++j) {
        int col = j * 16 + mrow;
        float h = acc[j][v];
        if (res) { h += xres[rb + col]; xres[rb + col] = h; }
        float xo = fmaxf(h, 0.f);
        x[rb + col] = xo;
        atomicAdd(&gpool[gofs + col], xo);
      }
    }
  }
}

// ---------------------------------------------------------------------------
// Post MLP + edit-distance head: out[g] = sum |post(gq[g]) - post(gc[g])|
// (relu(d)+relu(-d) == |d|).  Tiny: 128 graphs.
// ---------------------------------------------------------------------------
__global__ __launch_bounds__(64) void post_kernel(
    const float* __restrict__ gq, const float* __restrict__ gc,
    const float* __restrict__ w1, const float* __restrict__ b1,
    const float* __restrict__ w2, const float* __restrict__ b2,
    float* __restrict__ out) {
  int g = blockIdx.x;
  int t = threadIdx.x;
  __shared__ float yq[HID], yc[HID], partial[64];
  float aq = b1[t], ac = b1[t];
  for (int k = 0; k < POOLD; ++k) {
    float w = w1[k * HID + t];
    aq += gq[(size_t)g * POOLD + k] * w;
    ac += gc[(size_t)g * POOLD + k] * w;
  }
  yq[t] = fmaxf(aq, 0.f);
  yc[t] = fmaxf(ac, 0.f);
  __syncthreads();
  float contrib = 0.f;
  if (t < OUTDIM) {
    float zq = b2[t], zc = b2[t];
    for (int k = 0; k < HID; ++k) {
      float w = w2[k * OUTDIM + t];
      zq += yq[k] * w;
      zc += yc[k] * w;
    }
    float d = zq - zc;
    contrib = fmaxf(d, 0.f) + fmaxf(-d, 0.f);
  }
  partial[t] = contrib;
  __syncthreads();
  for (int s = 32; s > 0; s >>= 1) {
    if (t < s) partial[t] += partial[t + s];
    __syncthreads();
  }
  if (t == 0) out[g] = partial[0];
}

// ---------------------------------------------------------------------------
extern "C" void kernel_launch(void* const* d_in, const int* in_sizes, int n_in,
                              void* d_out, int out_size, void* d_ws,
                              size_t ws_size, hipStream_t stream) {
  const float* x_q     = (const float*)d_in[0];
  const int*   ei_q    = (const int*)d_in[1];
  const int*   batch_q = (const int*)d_in[2];
  const float* x_c     = (const float*)d_in[3];
  const int*   ei_c    = (const int*)d_in[4];
  const int*   batch_c = (const int*)d_in[5];
  const float* pre_w   = (const float*)d_in[6];
  const float* pre_b   = (const float*)d_in[7];
  const float* conv_w1 = (const float*)d_in[8];
  const float* conv_b1 = (const float*)d_in[9];
  const float* conv_w2 = (const float*)d_in[10];
  const float* conv_b2 = (const float*)d_in[11];
  const float* post_w1 = (const float*)d_in[12];
  const float* post_b1 = (const float*)d_in[13];
  const float* post_w2 = (const float*)d_in[14];
  const float* post_b2 = (const float*)d_in[15];

  float* ws   = (float*)d_ws;
  float* xcur = ws;                                  // N*64
  float* xres = xcur + (size_t)NNODES * HID;         // N*64
  float* agg  = xres + (size_t)NNODES * HID;         // N*64
  float* gq   = agg  + (size_t)NNODES * HID;         // 128*320
  float* gc   = gq + (size_t)NGRAPH * POOLD;         // 128*320

  (void)hipMemsetAsync(gq, 0, 2ull * NGRAPH * POOLD * sizeof(float), stream);

  const int tiles = (NNODES + 15) / 16;              // 6250
  const int mgrid = (tiles + 7) / 8;                 // 782 blocks x 256
  const int sgrid = (int)(((long long)NEDGES * 16 + 255) / 256);  // 100000

  for (int side = 0; side < 2; ++side) {
    const float* xi  = side ? x_c : x_q;
    const int*   ei  = side ? ei_c : ei_q;
    const int*   bat = side ? batch_c : batch_q;
    float*       gp  = side ? gc : gq;

    pre_kernel<<<mgrid, 256, 0, stream>>>(xi, pre_w, pre_b, bat, xcur, xres, gp);
    for (int l = 0; l < NLAYERS; ++l) {
      (void)hipMemsetAsync(agg, 0, (size_t)NNODES * HID * sizeof(float), stream);
      scatter_kernel<<<sgrid, 256, 0, stream>>>(xcur, ei, agg);
      layer_kernel<<<mgrid, 256, 0, stream>>>(xcur, agg, conv_w1, conv_b1,
                                              conv_w2, conv_b2, bat, xres, gp,
                                              l);
    }
  }
  post_kernel<<<NGRAPH, 64, 0, stream>>>(gq, gc, post_w1, post_b1, post_w2,
                                         post_b2, (float*)d_out);
}